// ContrastiveLoss_9809705304608
// MI455X (gfx1250) — compile-verified
//
#include <hip/hip_runtime.h>
#include <hip/hip_bf16.h>

// ---------------------------------------------------------------------------
// Discriminative (contrastive) clustering loss, MI455X / gfx1250.
//
// x   : [N=2, E=16, V=2^20] f32,  target : [N=2, V] int32 labels 0..15
//
// Memory-bound (136 MB unique, 2 passes, ~3 GFLOP). 136 MB < 192 MB L2 ->
// pass 2 is L2-resident; HBM floor ~6 us @ 23.3 TB/s. All loads RT-hinted
// (non-temporal would evict the L2-resident second pass).
//
// Pass 1 (dl_sums): segment-sum GEMM  sums[e][c] += x[e][v] * onehot[v][c]
//   via V_WMMA_F32_16X16X4_F32 (only f32 WMMA -> exact f32 sums). B128 loads:
//   each lane loads float4 of x and int4 of labels; 2 WMMAs cover 8 voxels
//   (voxel->K mapping chosen so A and the compare-built one-hot B agree).
// Pass 2 (dl_var): per-voxel hinge^2 distance to own mean; float4/int4
//   vectorized coalesced loads, conflict-free LDS mean lookup meansT[e][c]
//   (distinct labels -> distinct banks, equal labels -> broadcast),
//   ds_add_f32 label binning.
// Pass 3 (dl_final): Gram G = M*M^T of the 16x16 means via 4 chained f32
//   WMMAs (A == B), then distance/variance/reg terms -> scalar.
// ---------------------------------------------------------------------------

typedef __attribute__((ext_vector_type(2))) float v2f;
typedef __attribute__((ext_vector_type(8))) float v8f;

#define NB   2
#define CN   16
#define EN   16
#define VN   (64 * 128 * 128)   // 1048576 voxels per batch item

// workspace layout (floats):
//   [0,    512) sums   : n*256 + c*16 + e
//   [512,  544) counts : n*16 + c
//   [544,  576) var    : n*16 + c
#define WS_SUMS   0
#define WS_CNT    512
#define WS_VAR    544
#define WS_FLOATS 576

__global__ __launch_bounds__(576) void dl_zero(float* __restrict__ ws) {
    int t = threadIdx.x;
    if (t < WS_FLOATS) ws[t] = 0.0f;
}

// ---------------- Pass 1: segment sums via WMMA -----------------------------
// grid = 256 blocks (128 per n), block = 256 (8 waves). 1024 waves per n,
// 1024 voxels per wave, 8 voxels (2 WMMAs) per iteration -> 128 iterations.
__global__ __launch_bounds__(256) void dl_sums(const float* __restrict__ x,
                                               const int*   __restrict__ tgt,
                                               float*       __restrict__ ws) {
    const int lane = threadIdx.x & 31;
    const int wave = threadIdx.x >> 5;
    const int n    = blockIdx.x >> 7;          // 128 blocks per n
    const int blk  = blockIdx.x & 127;
    const int gw   = blk * 8 + wave;           // 0..1023 wave id within n
    const int l15  = lane & 15;                // A row (=e) and B col (=c)
    const int h4   = (lane >> 4) << 2;         // 0 (K=0,1) or 4 (K=2,3) voxels
    const int v0base = gw * (VN / 1024);       // 1024 voxels per wave

    const float* xrow = x + (size_t)n * EN * VN + (size_t)l15 * VN;
    const int*   trow = tgt + (size_t)n * VN;

    v8f acc = {};          // D tile: acc[j] = sums[e = j+8*(lane>=16)][c = l15]
    float cnt = 0.0f;

    const int iters = (VN / 1024) / 8;         // 128
    int v0 = v0base;
    for (int it = 0; it < iters; ++it, v0 += 8) {
        // A: x[e = l15][v0+h4 .. v0+h4+3] (16B aligned, fully line-packed).
        float4 va = *(const float4*)(xrow + v0 + h4);
        int4   lb = *(const int4*)(trow + v0 + h4);
        __builtin_prefetch(xrow + v0 + 64, 0, 3);

        // WMMA #1: voxels {v0+h4+0, v0+h4+1} at K = {2*half, 2*half+1}
        v2f a1, b1;
        a1.x = va.x; a1.y = va.y;
        b1.x = (lb.x == l15) ? 1.0f : 0.0f;
        b1.y = (lb.y == l15) ? 1.0f : 0.0f;
        acc = __builtin_amdgcn_wmma_f32_16x16x4_f32(
                  false, a1, false, b1, (short)0, acc, false, false);

        // WMMA #2: voxels {v0+h4+2, v0+h4+3} at the same K slots
        v2f a2, b2;
        a2.x = va.z; a2.y = va.w;
        b2.x = (lb.z == l15) ? 1.0f : 0.0f;
        b2.y = (lb.w == l15) ? 1.0f : 0.0f;
        acc = __builtin_amdgcn_wmma_f32_16x16x4_f32(
                  false, a2, false, b2, (short)0, acc, false, false);

        cnt += b1.x + b1.y + b2.x + b2.y;      // 8 ones per wave-iteration
    }

    // flush partials: counts (both half-waves contribute) + 16x16 D tile
    atomicAdd(&ws[WS_CNT + n * 16 + l15], cnt);
#pragma unroll
    for (int j = 0; j < 8; ++j) {
        int eo = j + ((lane >> 4) << 3);       // D row -> embedding index
        atomicAdd(&ws[WS_SUMS + n * 256 + l15 * 16 + eo], acc[j]);
    }
}

// ---------------- Pass 2: variance term -------------------------------------
// grid = 1024 blocks (512 per n), block = 256; 2048 voxels per block,
// 4 voxels per thread per iteration (float4/int4), 2 iterations.
__global__ __launch_bounds__(256) void dl_var(const float* __restrict__ x,
                                              const int*   __restrict__ tgt,
                                              float*       __restrict__ ws) {
    __shared__ float meansT[EN * CN];          // [e][c]: conflict-free lookup
    __shared__ float vloc[CN];

    const int tid = threadIdx.x;
    const int n   = blockIdx.x >> 9;           // 512 blocks per n
    const int blk = blockIdx.x & 511;

    if (tid < 256) {
        int c = tid >> 4, e = tid & 15;
        float m = ws[WS_SUMS + n * 256 + c * 16 + e] /
                  ws[WS_CNT + n * 16 + c];
        meansT[e * 16 + c] = m;
    }
    if (tid < CN) vloc[tid] = 0.0f;
    __syncthreads();

    const size_t base  = (size_t)n * EN * VN;
    const int    vbase = blk * (VN / 512);     // 2048 voxels per block

#pragma unroll 1
    for (int i = 0; i < 2; ++i) {              // 2 x (256 threads x 4 voxels)
        int v = vbase + (i * 256 + tid) * 4;
        int4 lb = *(const int4*)(tgt + (size_t)n * VN + v);
        float nqx = 0.0f, nqy = 0.0f, nqz = 0.0f, nqw = 0.0f;
#pragma unroll
        for (int e = 0; e < EN; ++e) {
            float4 xa = *(const float4*)(x + base + (size_t)e * VN + v);
            float dx = xa.x - meansT[e * 16 + lb.x];
            float dy = xa.y - meansT[e * 16 + lb.y];
            float dz = xa.z - meansT[e * 16 + lb.z];
            float dw = xa.w - meansT[e * 16 + lb.w];
            nqx = fmaf(dx, dx, nqx);
            nqy = fmaf(dy, dy, nqy);
            nqz = fmaf(dz, dz, nqz);
            nqw = fmaf(dw, dw, nqw);
        }
        float hx = sqrtf(nqx) - 0.5f;          // DELTA_VAR
        float hy = sqrtf(nqy) - 0.5f;
        float hz = sqrtf(nqz) - 0.5f;
        float hw = sqrtf(nqw) - 0.5f;
        hx = (hx > 0.0f) ? hx * hx : 0.0f;
        hy = (hy > 0.0f) ? hy * hy : 0.0f;
        hz = (hz > 0.0f) ? hz * hz : 0.0f;
        hw = (hw > 0.0f) ? hw * hw : 0.0f;
        atomicAdd(&vloc[lb.x], hx);            // ds_add_f32
        atomicAdd(&vloc[lb.y], hy);
        atomicAdd(&vloc[lb.z], hz);
        atomicAdd(&vloc[lb.w], hw);
    }
    __syncthreads();
    if (tid < CN) atomicAdd(&ws[WS_VAR + n * 16 + tid], vloc[tid]);
}

// ---------------- Pass 3: distance / variance / reg terms -------------------
// One wave. Gram matrix of means via 4 chained f32 WMMAs (A == B).
__global__ __launch_bounds__(32) void dl_final(const float* __restrict__ ws,
                                               float*       __restrict__ out) {
    __shared__ float means_l[CN * EN];         // [c][e]
    __shared__ float r_l[CN];                  // ||mean_c||^2
    __shared__ float red[96];

    const int lane = threadIdx.x;
    const int l15  = lane & 15;
    float total = 0.0f;

    for (int n = 0; n < NB; ++n) {
#pragma unroll
        for (int j = 0; j < 8; ++j) {          // 32 lanes * 8 = 256 means
            int idx = lane * 8 + j;
            int c = idx >> 4, e = idx & 15;
            means_l[idx] = ws[WS_SUMS + n * 256 + c * 16 + e] /
                           ws[WS_CNT + n * 16 + c];
        }
        __syncthreads();
        if (lane < CN) {
            float s = 0.0f;
            for (int e = 0; e < EN; ++e) {
                float m = means_l[lane * 16 + e];
                s = fmaf(m, m, s);
            }
            r_l[lane] = s;
        }
        __syncthreads();

        // Gram: G[c1][c2] = sum_e mean[c1][e]*mean[c2][e]; K=16 in 4 steps.
        v8f g = {};
#pragma unroll
        for (int s = 0; s < 4; ++s) {
            int kb = s * 4 + ((lane >> 4) << 1);
            v2f a;
            a.x = means_l[l15 * 16 + kb];
            a.y = means_l[l15 * 16 + kb + 1];
            g = __builtin_amdgcn_wmma_f32_16x16x4_f32(
                    false, a, false, a, (short)0, g, false, false);
        }

        // distance term: hinge( 2*DELTA_DIST - ||mu_i - mu_j|| )^2 off-diag
        float dpart = 0.0f;
#pragma unroll
        for (int j = 0; j < 8; ++j) {
            int c1 = j + ((lane >> 4) << 3);
            int c2 = l15;
            if (c1 != c2) {
                float d2 = r_l[c1] + r_l[c2] - 2.0f * g[j];
                d2 = (d2 > 0.0f) ? d2 : 0.0f;
                float h = 3.0f - sqrtf(d2);    // 2*DELTA_DIST = 3.0
                if (h > 0.0f) dpart = fmaf(h, h, dpart);
            }
        }
        red[lane]      = dpart;
        red[32 + lane] = (lane < CN)
                       ? (ws[WS_VAR + n * 16 + lane] / ws[WS_CNT + n * 16 + lane])
                       : 0.0f;
        red[64 + lane] = (lane < CN) ? sqrtf(r_l[lane]) : 0.0f;
        __syncthreads();

        if (lane == 0) {
            float ds = 0.0f, vs = 0.0f, rs = 0.0f;
            for (int i = 0; i < 32; ++i) {
                ds += red[i];
                vs += red[32 + i];
                rs += red[64 + i];
            }
            // ALPHA*var + BETA*dist + GAMMA*reg
            total += vs / 16.0f + ds / 240.0f + 0.001f * (rs / 16.0f);
        }
        __syncthreads();                       // protect means_l for next n
    }
    if (lane == 0) out[0] = total / (float)NB;
}

extern "C" void kernel_launch(void* const* d_in, const int* in_sizes, int n_in,
                              void* d_out, int out_size, void* d_ws, size_t ws_size,
                              hipStream_t stream) {
    const float* x   = (const float*)d_in[0];  // [2,16,64,128,128] f32
    const int*   tgt = (const int*)d_in[1];    // [2,64,128,128] i32
    float* out = (float*)d_out;
    float* ws  = (float*)d_ws;                 // needs 576 floats (2304 B)

    dl_zero <<<1,    576, 0, stream>>>(ws);
    dl_sums <<<256,  256, 0, stream>>>(x, tgt, ws);
    dl_var  <<<1024, 256, 0, stream>>>(x, tgt, ws);
    dl_final<<<1,    32,  0, stream>>>(ws, out);
}